// VTFfeatures_34815004901659
// MI455X (gfx1250) — compile-verified
//
#include <hip/hip_runtime.h>
#include <cmath>

// ---------------------------------------------------------------------------
// VTF cross-modal attention for MI455X (gfx1250), f32 end-to-end.
// GEMMs on V_WMMA_F32_16X16X4_F32; dominant GEMM (v @ att^T) double-buffers
// LDS tiles via the Tensor Data Mover (TENSOR_LOAD_TO_LDS + s_wait_tensorcnt),
// overlapping DMA of chunk i+1 with WMMA compute of chunk i.
// ---------------------------------------------------------------------------

typedef float v2f __attribute__((ext_vector_type(2)));
typedef float v8f __attribute__((ext_vector_type(8)));
typedef unsigned int u32x4 __attribute__((ext_vector_type(4)));
typedef int i32x4 __attribute__((ext_vector_type(4)));
typedef int i32x8 __attribute__((ext_vector_type(8)));

#define BT   12            // B*T
#define CF   1024          // fused channels
#define CQ   128           // q/k channels
#define NV   2401          // valid spatial positions (49*49)
#define NP   2432          // padded N = 38*64
#define HW   49            // 7*7 per-source spatial

#define KC     128         // K-chunk (m) staged per LDS tile
#define NCH    (NP / KC)   // 19 chunks
#define LDSROW 130         // padded LDS row stride in floats (bank spread)
#define VOFFW  (64 * LDSROW)          // float offset of V tile inside a buffer
#define BUFW   (2 * 64 * LDSROW)      // floats per double-buffer slot (att+v)

// workspace layout (floats)
#define SX   ((size_t)BT * CF * NP)
#define SQ   ((size_t)BT * CQ * NP)

#if defined(__AMDGCN__) && defined(__has_builtin)
#if __has_builtin(__builtin_amdgcn_tensor_load_to_lds) && \
    __has_builtin(__builtin_amdgcn_s_wait_tensorcnt)
#define HAVE_TDM 1
#endif
#endif
#ifndef HAVE_TDM
#define HAVE_TDM 0
#endif

#if HAVE_TDM
// Issue a TDM load of a 2D f32 tile (rows x cols), global row stride NP
// elements, into LDS at `lds` with +2-DWORD padding every 128 DWORDs
// (LDS row stride becomes LDSROW floats). The LDS byte address is the low
// 32 bits of the generic pointer;
// deriving it from the real pointer also makes the write visible to the
// compiler so subsequent ds_loads are not folded away.
__device__ __forceinline__ void tdm_load_tile(const float* g, const float* lds,
                                              int cols, int rows) {
  unsigned long long ga = (unsigned long long)g;
  unsigned lds_off = (unsigned)(unsigned long long)(uintptr_t)lds;
  u32x4 g0;
  g0[0] = 1u;                                   // count=1 (valid), user mode
  g0[1] = lds_off;                              // LDS byte address
  g0[2] = (unsigned)ga;                         // global addr [31:0]
  g0[3] = ((unsigned)(ga >> 32) & 0x01FFFFFFu)  // global addr [56:32]
          | 0x80000000u;                        // type=2 ("image")
  i32x8 g1;
  g1[0] = 0x00020000                            // data_size = 4B
        | (1 << 20)                             // pad_enable
        | (6 << 22)                             // pad_interval: 128 DWORDs
        | (1 << 25);                            // pad_amount: 2 DWORDs
  g1[1] = (int)(((unsigned)NP & 0xFFFFu) << 16);            // tensor_dim0 lo16
  g1[2] = (int)((((unsigned)NP >> 16) & 0xFFFFu)            // tensor_dim0 hi16
        | (((unsigned)NP & 0xFFFFu) << 16));                // tensor_dim1 lo16
  g1[3] = (int)((((unsigned)NP >> 16) & 0xFFFFu))           // tensor_dim1 hi16
        | (cols << 16);                                     // tile_dim0
  g1[4] = rows;                                             // tile_dim1 (dim2=0)
  g1[5] = NP;                                               // dim0_stride lo32
  g1[6] = 0;                                                // stride hi / dim1_stride
  g1[7] = 0;
  i32x4 z4 = {0, 0, 0, 0};
#if __clang_major__ >= 23
  i32x8 z8 = {0, 0, 0, 0, 0, 0, 0, 0};
  __builtin_amdgcn_tensor_load_to_lds(g0, g1, z4, z4, z8, 0);
#else
  __builtin_amdgcn_tensor_load_to_lds(g0, g1, z4, z4, 0);
#endif
}
#endif  // HAVE_TDM

// ---------------------------------------------------------------------------
// 1) Build fused feature map x[b][c][n], zero-padded to NP.
// ---------------------------------------------------------------------------
__global__ void build_x_kernel(const float* __restrict__ img,
                               const float* __restrict__ tac,
                               float* __restrict__ x) {
  size_t idx = (size_t)blockIdx.x * blockDim.x + threadIdx.x;
  const size_t total = SX;
  if (idx >= total) return;
  int n = (int)(idx % NP);
  int c = (int)((idx / NP) % CF);
  int b = (int)(idx / ((size_t)NP * CF));
  float val = 0.0f;
  if (n < NV) {
    int i = n / 49, j = n - 49 * (n / 49);
    if (c < 512)
      val = tac[((size_t)b * 512 + c) * HW + (i / 7) * 7 + (j / 7)];
    else
      val = img[((size_t)b * 512 + (c - 512)) * HW + (i % 7) * 7 + (j % 7)];
  }
  x[idx] = val;
}

// ---------------------------------------------------------------------------
// 2) Projection GEMM:  y[b][Co][NP] = W[Co][CF] * x[b][CF][NP] + bias
// ---------------------------------------------------------------------------
__global__ void gemm_proj_kernel(const float* __restrict__ W,
                                 const float* __restrict__ bias,
                                 const float* __restrict__ x,
                                 float* __restrict__ y, int Co) {
  const int lane = threadIdx.x & 31;
  const int wave = threadIdx.x >> 5;
  const int wid  = blockIdx.x * (blockDim.x >> 5) + wave;
  const int tilesO  = Co >> 4;
  const int groupsN = NP / 64;
  const int perB = tilesO * groupsN;
  const int b = wid / perB;
  if (b >= BT) return;
  int rem = wid - b * perB;
  const int obase = (rem / groupsN) * 16;
  const int nbase = (rem % groupsN) * 64;
  const int ml = lane & 15;
  const int kh = lane >> 4;

  const float* xb = x + (size_t)b * CF * NP;
  float*       yb = y + (size_t)b * Co * NP;

  v8f acc0 = {}, acc1 = {}, acc2 = {}, acc3 = {};
  for (int k0 = 0; k0 < CF; k0 += 4) {
    const int ka = k0 + 2 * kh;
    v2f A;
    A.x = W[(size_t)(obase + ml) * CF + ka];
    A.y = W[(size_t)(obase + ml) * CF + ka + 1];
    const float* xr0 = xb + (size_t)ka * NP;
    const float* xr1 = xr0 + NP;
    v2f B0 = { xr0[nbase + ml],      xr1[nbase + ml]      };
    v2f B1 = { xr0[nbase + 16 + ml], xr1[nbase + 16 + ml] };
    v2f B2 = { xr0[nbase + 32 + ml], xr1[nbase + 32 + ml] };
    v2f B3 = { xr0[nbase + 48 + ml], xr1[nbase + 48 + ml] };
    acc0 = __builtin_amdgcn_wmma_f32_16x16x4_f32(false, A, false, B0, (short)0, acc0, false, false);
    acc1 = __builtin_amdgcn_wmma_f32_16x16x4_f32(false, A, false, B1, (short)0, acc1, false, false);
    acc2 = __builtin_amdgcn_wmma_f32_16x16x4_f32(false, A, false, B2, (short)0, acc2, false, false);
    acc3 = __builtin_amdgcn_wmma_f32_16x16x4_f32(false, A, false, B3, (short)0, acc3, false, false);
  }
  const int rbase = obase + (kh ? 8 : 0);
  for (int r = 0; r < 8; ++r) {
    float bv = bias[rbase + r];
    float* yr = yb + (size_t)(rbase + r) * NP + nbase + ml;
    yr[0]  = acc0[r] + bv;
    yr[16] = acc1[r] + bv;
    yr[32] = acc2[r] + bv;
    yr[48] = acc3[r] + bv;
  }
}

// ---------------------------------------------------------------------------
// 3) Attention scores: att[b][n][m] = sum_o qT[o][n] * k[o][m]
// ---------------------------------------------------------------------------
__global__ void gemm_att_kernel(const float* __restrict__ qT,
                                const float* __restrict__ kM,
                                float* __restrict__ att) {
  const int lane = threadIdx.x & 31;
  const int wave = threadIdx.x >> 5;
  const int wid  = blockIdx.x * (blockDim.x >> 5) + wave;
  const int tilesN  = NP / 16;
  const int groupsM = NP / 64;
  const int perB = tilesN * groupsM;
  const int b = wid / perB;
  if (b >= BT) return;
  int rem = wid - b * perB;
  const int nbase = (rem / groupsM) * 16;
  const int mbase = (rem % groupsM) * 64;
  const int ml = lane & 15;
  const int kh = lane >> 4;

  const float* qb = qT + (size_t)b * CQ * NP;
  const float* kb = kM + (size_t)b * CQ * NP;
  float*       ab = att + (size_t)b * NP * NP;

  v8f acc0 = {}, acc1 = {}, acc2 = {}, acc3 = {};
  for (int k0 = 0; k0 < CQ; k0 += 4) {
    const int ka = k0 + 2 * kh;
    v2f A;
    A.x = qb[(size_t)ka * NP + nbase + ml];
    A.y = qb[(size_t)(ka + 1) * NP + nbase + ml];
    const float* kr0 = kb + (size_t)ka * NP;
    const float* kr1 = kr0 + NP;
    v2f B0 = { kr0[mbase + ml],      kr1[mbase + ml]      };
    v2f B1 = { kr0[mbase + 16 + ml], kr1[mbase + 16 + ml] };
    v2f B2 = { kr0[mbase + 32 + ml], kr1[mbase + 32 + ml] };
    v2f B3 = { kr0[mbase + 48 + ml], kr1[mbase + 48 + ml] };
    acc0 = __builtin_amdgcn_wmma_f32_16x16x4_f32(false, A, false, B0, (short)0, acc0, false, false);
    acc1 = __builtin_amdgcn_wmma_f32_16x16x4_f32(false, A, false, B1, (short)0, acc1, false, false);
    acc2 = __builtin_amdgcn_wmma_f32_16x16x4_f32(false, A, false, B2, (short)0, acc2, false, false);
    acc3 = __builtin_amdgcn_wmma_f32_16x16x4_f32(false, A, false, B3, (short)0, acc3, false, false);
  }
  const int rbase = nbase + (kh ? 8 : 0);
  for (int r = 0; r < 8; ++r) {
    float* ar = ab + (size_t)(rbase + r) * NP + mbase + ml;
    ar[0]  = acc0[r];
    ar[16] = acc1[r];
    ar[32] = acc2[r];
    ar[48] = acc3[r];
  }
}

// ---------------------------------------------------------------------------
// 4) Row softmax over valid m < NV; padded columns -> 0.
// ---------------------------------------------------------------------------
__global__ void softmax_kernel(float* __restrict__ att) {
  const int row = blockIdx.x;
  const int b = row / NV;
  const int n = row - b * NV;
  float* p = att + ((size_t)b * NP + n) * NP;
  const int tid  = threadIdx.x;
  const int lane = tid & 31;
  const int wv   = tid >> 5;
  __shared__ float sred[8];
  __shared__ float sb_max, sb_sum;

  float mx = -3.4e38f;
  for (int m = tid; m < NV; m += blockDim.x) mx = fmaxf(mx, p[m]);
  for (int off = 16; off; off >>= 1) mx = fmaxf(mx, __shfl_xor(mx, off, 32));
  if (lane == 0) sred[wv] = mx;
  __syncthreads();
  if (tid == 0) {
    float v = sred[0];
    for (int i = 1; i < 8; ++i) v = fmaxf(v, sred[i]);
    sb_max = v;
  }
  __syncthreads();
  mx = sb_max;

  float s = 0.0f;
  for (int m = tid; m < NV; m += blockDim.x) s += __expf(p[m] - mx);
  for (int off = 16; off; off >>= 1) s += __shfl_xor(s, off, 32);
  __syncthreads();
  if (lane == 0) sred[wv] = s;
  __syncthreads();
  if (tid == 0) {
    float v = 0.0f;
    for (int i = 0; i < 8; ++i) v += sred[i];
    sb_sum = v;
  }
  __syncthreads();
  const float inv = 1.0f / sb_sum;

  for (int m = tid; m < NP; m += blockDim.x)
    p[m] = (m < NV) ? __expf(p[m] - mx) * inv : 0.0f;
}

// ---------------------------------------------------------------------------
// 5) out[b][c][n] = gamma * sum_m v[c][m] * att[n][m] + x[c][n]
//    Block = 4 waves -> 64(c) x 64(n) tile. K chunked by 128. att & v chunks
//    are TDM double-buffered in LDS: wave 0 issues chunk i+1's two descriptors
//    right after the barrier, all waves compute chunk i from the other buffer,
//    s_wait_tensorcnt(0) at the next iteration top covers the in-flight pair.
// ---------------------------------------------------------------------------
__global__ void gemm_out_kernel(const float* __restrict__ vM,
                                const float* __restrict__ att,
                                const float* __restrict__ x,
                                const float* __restrict__ gamma,
                                float* __restrict__ out) {
  __shared__ float smem[2 * BUFW];              // [buf0: att|v][buf1: att|v]
  const int lane = threadIdx.x & 31;
  const int wave = threadIdx.x >> 5;
  const int groupsN = NP / 64;                  // 38
  const int groupsC = CF / 64;                  // 16
  const int perB = groupsC * groupsN;
  const int b = blockIdx.x / perB;
  if (b >= BT) return;
  int rem = blockIdx.x - b * perB;
  const int cblk = (rem / groupsN) * 64;
  const int nblk = (rem % groupsN) * 64;
  const int ml = lane & 15;
  const int kh = lane >> 4;

  const float* vb = vM  + (size_t)b * CF * NP;
  const float* ab = att + (size_t)b * NP * NP;

#if HAVE_TDM
  if (wave == 0) {                              // prologue: stage chunk 0
    tdm_load_tile(ab + (size_t)nblk * NP, smem,         KC, 64);
    tdm_load_tile(vb + (size_t)cblk * NP, smem + VOFFW, KC, 64);
  }
#endif

  v8f acc0 = {}, acc1 = {}, acc2 = {}, acc3 = {};
  for (int i = 0; i < NCH; ++i) {
    float* base = smem + (i & 1) * BUFW;
#if HAVE_TDM
    if (wave == 0) __builtin_amdgcn_s_wait_tensorcnt(0);  // chunk i resident
    __syncthreads();                            // release tile i; tile i-1 free
    asm volatile("" ::: "memory");              // keep LDS reads materialized
    if (wave == 0 && i + 1 < NCH) {             // overlap DMA(i+1) with compute
      const size_t kn = (size_t)(i + 1) * KC;
      float* nbuf = smem + ((i + 1) & 1) * BUFW;
      tdm_load_tile(ab + (size_t)nblk * NP + kn, nbuf,         KC, 64);
      tdm_load_tile(vb + (size_t)cblk * NP + kn, nbuf + VOFFW, KC, 64);
    }
#else
    __syncthreads();
    for (int idx = threadIdx.x; idx < 64 * KC; idx += blockDim.x) {
      int r = idx >> 7, cix = idx & (KC - 1);   // KC == 128
      base[r * LDSROW + cix]         = ab[(size_t)(nblk + r) * NP + i * KC + cix];
      base[VOFFW + r * LDSROW + cix] = vb[(size_t)(cblk + r) * NP + i * KC + cix];
    }
    __syncthreads();
#endif
    const float* sa0 = base + (size_t)ml * LDSROW;          // att rows  0+ml
    const float* sa1 = sa0 + 16 * LDSROW;                   //          16+ml
    const float* sa2 = sa0 + 32 * LDSROW;                   //          32+ml
    const float* sa3 = sa0 + 48 * LDSROW;                   //          48+ml
    const float* sv  = base + VOFFW + (size_t)(wave * 16 + ml) * LDSROW;

    for (int ks = 0; ks < KC; ks += 4) {
      const int kk = ks + 2 * kh;
      v2f A  = { sv[kk],  sv[kk + 1]  };
      v2f B0 = { sa0[kk], sa0[kk + 1] };
      v2f B1 = { sa1[kk], sa1[kk + 1] };
      v2f B2 = { sa2[kk], sa2[kk + 1] };
      v2f B3 = { sa3[kk], sa3[kk + 1] };
      acc0 = __builtin_amdgcn_wmma_f32_16x16x4_f32(false, A, false, B0, (short)0, acc0, false, false);
      acc1 = __builtin_amdgcn_wmma_f32_16x16x4_f32(false, A, false, B1, (short)0, acc1, false, false);
      acc2 = __builtin_amdgcn_wmma_f32_16x16x4_f32(false, A, false, B2, (short)0, acc2, false, false);
      acc3 = __builtin_amdgcn_wmma_f32_16x16x4_f32(false, A, false, B3, (short)0, acc3, false, false);
    }
  }

  const float g = gamma[0];
  const int cbase = cblk + wave * 16 + (kh ? 8 : 0);
  const float* xb = x + (size_t)b * CF * NP;
  float* ob = out + (size_t)b * CF * NV;
  for (int r = 0; r < 8; ++r) {
    const int c = cbase + r;
    const float* xr = xb + (size_t)c * NP;
    float* orow = ob + (size_t)c * NV;
    int n0 = nblk + ml;
    if (n0      < NV) orow[n0]      = g * acc0[r] + xr[n0];
    if (n0 + 16 < NV) orow[n0 + 16] = g * acc1[r] + xr[n0 + 16];
    if (n0 + 32 < NV) orow[n0 + 32] = g * acc2[r] + xr[n0 + 32];
    if (n0 + 48 < NV) orow[n0 + 48] = g * acc3[r] + xr[n0 + 48];
  }
}

// ---------------------------------------------------------------------------
extern "C" void kernel_launch(void* const* d_in, const int* in_sizes, int n_in,
                              void* d_out, int out_size, void* d_ws, size_t ws_size,
                              hipStream_t stream) {
  const float* img   = (const float*)d_in[0];
  const float* tac   = (const float*)d_in[1];
  const float* Wq    = (const float*)d_in[2];
  const float* bq    = (const float*)d_in[3];
  const float* Wk    = (const float*)d_in[4];
  const float* bk    = (const float*)d_in[5];
  const float* Wv    = (const float*)d_in[6];
  const float* bv    = (const float*)d_in[7];
  const float* gamma = (const float*)d_in[8];
  float* out = (float*)d_out;

  float* ws  = (float*)d_ws;
  float* X   = ws;                   // [BT][CF][NP]
  float* V   = ws + SX;              // [BT][CF][NP]
  float* QT  = ws + 2 * SX;          // [BT][CQ][NP]
  float* K   = ws + 2 * SX + SQ;     // [BT][CQ][NP]
  float* ATT = ws + 2 * SX + 2 * SQ; // [BT][NP][NP]

  {
    size_t total = SX;
    int blocks = (int)((total + 255) / 256);
    build_x_kernel<<<blocks, 256, 0, stream>>>(img, tac, X);
  }
  {
    int wavesQK = BT * (CQ / 16) * (NP / 64);
    gemm_proj_kernel<<<wavesQK / 4, 128, 0, stream>>>(Wq, bq, X, QT, CQ);
    gemm_proj_kernel<<<wavesQK / 4, 128, 0, stream>>>(Wk, bk, X, K,  CQ);
    int wavesV = BT * (CF / 16) * (NP / 64);
    gemm_proj_kernel<<<wavesV / 4, 128, 0, stream>>>(Wv, bv, X, V, CF);
  }
  {
    int waves = BT * (NP / 16) * (NP / 64);
    gemm_att_kernel<<<waves / 4, 128, 0, stream>>>(QT, K, ATT);
  }
  softmax_kernel<<<BT * NV, 256, 0, stream>>>(ATT);
  {
    int blocks = BT * (CF / 64) * (NP / 64);   // 7296: one 64x64 tile per block
    gemm_out_kernel<<<blocks, 128, 0, stream>>>(V, ATT, X, gamma, out);
  }
}